// Net_46016279609830
// MI455X (gfx1250) — compile-verified
//
#include <hip/hip_runtime.h>
#include <math.h>

typedef float v2f __attribute__((ext_vector_type(2)));
typedef float v8f __attribute__((ext_vector_type(8)));

#define F_IN 128
#define F_H  16
#define G_WAVES 4   // waves (16-row tiles) per block in GEMM kernels

// ---------------------------------------------------------------------------
// Degree / normalization
// ---------------------------------------------------------------------------
__global__ void k_init_deg(float* __restrict__ deg, int N) {
  int i = blockIdx.x * blockDim.x + threadIdx.x;
  if (i < N) deg[i] = 1.0f;              // self-loop contribution
}

__global__ void k_edge_deg(const int* __restrict__ dst, float* __restrict__ deg, int E) {
  int i = blockIdx.x * blockDim.x + threadIdx.x;
  if (i < E) atomicAdd(&deg[dst[i]], 1.0f);
}

__global__ void k_rsqrt(float* __restrict__ deg, int N) {
  int i = blockIdx.x * blockDim.x + threadIdx.x;
  if (i < N) deg[i] = rsqrtf(deg[i]);    // deg >= 1 always (self-loops)
}

// ---------------------------------------------------------------------------
// GEMM1: H = X (N x 128) @ W1 (128 x 16) via V_WMMA_F32_16X16X4_F32
// Epilogue: H -> bufA, AGG(self-loop init) = H * dinv^2 -> bufB
// One wave computes one 16x16 output tile; K loop = 128/4 = 32 WMMAs.
// ---------------------------------------------------------------------------
__global__ void __launch_bounds__(32 * G_WAVES)
k_gemm1_wmma(const float* __restrict__ X, const float* __restrict__ W,
             const float* __restrict__ dinv,
             float* __restrict__ H, float* __restrict__ AGG, int N) {
  __shared__ float lw[F_IN * F_H];                         // 8 KB
  __shared__ __attribute__((aligned(16))) float la[G_WAVES][16][F_IN + 4]; // pad->132: no bank conflicts

  const int wave = threadIdx.x >> 5;
  const int lane = threadIdx.x & 31;
  const int tile0 = (blockIdx.x * G_WAVES + wave) * 16;

  // cooperative load of W1 (128x16 row-major)
  for (int i = threadIdx.x; i < F_IN * F_H; i += blockDim.x) lw[i] = W[i];

  // wave loads its 16x128 X tile, coalesced float4 (512 float4 / 32 lanes = 16 iters)
  for (int it = 0; it < 16; ++it) {
    int idx = it * 32 + lane;          // 0..511
    int r   = idx >> 5;                // row within tile (32 float4 per row)
    int c4  = idx & 31;                // float4 column
    int row = tile0 + r;
    float4 v = make_float4(0.f, 0.f, 0.f, 0.f);
    if (row < N) v = ((const float4*)X)[(long long)row * 32 + c4];
    *(float4*)&la[wave][r][c4 * 4] = v;
  }
  __syncthreads();

  const int m    = lane & 15;          // A row (M) == B col (N) for this lane
  const int koff = (lane >> 4) << 1;   // 0 for lanes 0-15, 2 for lanes 16-31

  v8f c = {0.f, 0.f, 0.f, 0.f, 0.f, 0.f, 0.f, 0.f};
  #pragma unroll
  for (int kb = 0; kb < F_IN / 4; ++kb) {
    int k = kb * 4 + koff;
    v2f a = *(const v2f*)&la[wave][m][k];                 // A: M=m, K={k,k+1}
    v2f b = { lw[k * F_H + m], lw[(k + 1) * F_H + m] };   // B: K={k,k+1}, N=m
    c = __builtin_amdgcn_wmma_f32_16x16x4_f32(false, a, false, b, (short)0, c,
                                              false, false);
  }

  // store: VGPR j holds M = j (lanes 0-15) / M = j+8 (lanes 16-31), N = lane&15
  const int half = lane >> 4;
  #pragma unroll
  for (int j = 0; j < 8; ++j) {
    int row = tile0 + j + half * 8;
    if (row < N) {
      float v  = c[j];
      float di = dinv[row];
      long long o = (long long)row * F_H + m;
      H[o]   = v;
      AGG[o] = v * di * di;            // self-loop message, norm = dinv^2
    }
  }
}

// ---------------------------------------------------------------------------
// GEMM2: H2 = A1 (N x 16) @ W2 (16 x 16), K = 16 -> 4 WMMAs.
// Epilogue: H2 -> bufA, OUT(self-loop init) = H2 * dinv^2 -> d_out
// ---------------------------------------------------------------------------
__global__ void __launch_bounds__(32 * G_WAVES)
k_gemm2_wmma(const float* __restrict__ A1, const float* __restrict__ W,
             const float* __restrict__ dinv,
             float* __restrict__ H2, float* __restrict__ OUT, int N) {
  __shared__ float lw[F_H * F_H];
  const int wave = threadIdx.x >> 5;
  const int lane = threadIdx.x & 31;
  const int tile0 = (blockIdx.x * G_WAVES + wave) * 16;

  for (int i = threadIdx.x; i < F_H * F_H; i += blockDim.x) lw[i] = W[i];
  __syncthreads();

  const int m    = lane & 15;
  const int koff = (lane >> 4) << 1;
  const int rowm = min(tile0 + m, N - 1);   // clamp; OOB rows recompute row N-1, stores guarded

  v8f c = {0.f, 0.f, 0.f, 0.f, 0.f, 0.f, 0.f, 0.f};
  #pragma unroll
  for (int kb = 0; kb < F_H / 4; ++kb) {
    int k = kb * 4 + koff;
    v2f a = *(const v2f*)&A1[(long long)rowm * F_H + k];
    v2f b = { lw[k * F_H + m], lw[(k + 1) * F_H + m] };
    c = __builtin_amdgcn_wmma_f32_16x16x4_f32(false, a, false, b, (short)0, c,
                                              false, false);
  }

  const int half = lane >> 4;
  #pragma unroll
  for (int j = 0; j < 8; ++j) {
    int row = tile0 + j + half * 8;
    if (row < N) {
      float v  = c[j];
      float di = dinv[row];
      long long o = (long long)row * F_H + m;
      H2[o]  = v;
      OUT[o] = v * di * di;
    }
  }
}

// ---------------------------------------------------------------------------
// Edge scatter: 16 lanes per edge (one per feature) -> coalesced 64B gather of
// h[src] and 16 consecutive global_atomic_add_f32 into agg[dst].
// ---------------------------------------------------------------------------
__global__ void k_scatter(const float* __restrict__ h, const int* __restrict__ src,
                          const int* __restrict__ dst, const float* __restrict__ dinv,
                          float* __restrict__ agg, int total /* = E*16 */) {
  int t = blockIdx.x * blockDim.x + threadIdx.x;
  if (t >= total) return;
  int e = t >> 4;
  int f = t & 15;
  int s = src[e];
  int d = dst[e];
  float norm = dinv[s] * dinv[d];
  atomicAdd(&agg[(long long)d * F_H + f], h[(long long)s * F_H + f] * norm);
}

// ---------------------------------------------------------------------------
// a = relu(a + b1[f])
// ---------------------------------------------------------------------------
__global__ void k_bias_relu(float* __restrict__ a, const float* __restrict__ b, int total) {
  int i = blockIdx.x * blockDim.x + threadIdx.x;
  if (i < total) a[i] = fmaxf(a[i] + b[i & 15], 0.0f);
}

// ---------------------------------------------------------------------------
// out = log_softmax(out + b2) per row of 16, one thread per row (64B chunks)
// ---------------------------------------------------------------------------
__global__ void k_bias_logsoftmax(float* __restrict__ out, const float* __restrict__ b, int N) {
  int i = blockIdx.x * blockDim.x + threadIdx.x;
  if (i >= N) return;
  float* row = out + (long long)i * F_H;
  float v[F_H];
  #pragma unroll
  for (int f = 0; f < F_H; ++f) v[f] = row[f] + b[f];
  float m = v[0];
  #pragma unroll
  for (int f = 1; f < F_H; ++f) m = fmaxf(m, v[f]);
  float s = 0.f;
  #pragma unroll
  for (int f = 0; f < F_H; ++f) s += expf(v[f] - m);
  float l = m + logf(s);
  #pragma unroll
  for (int f = 0; f < F_H; ++f) row[f] = v[f] - l;
}

// ---------------------------------------------------------------------------
extern "C" void kernel_launch(void* const* d_in, const int* in_sizes, int n_in,
                              void* d_out, int out_size, void* d_ws, size_t ws_size,
                              hipStream_t stream) {
  const float* x   = (const float*)d_in[0];
  const int*  eidx = (const int*)d_in[1];
  const float* W1  = (const float*)d_in[2];
  const float* b1  = (const float*)d_in[3];
  const float* W2  = (const float*)d_in[4];
  const float* b2  = (const float*)d_in[5];
  float* out = (float*)d_out;

  const int N = in_sizes[0] / F_IN;
  const int E = in_sizes[1] / 2;
  const int* src = eidx;          // edge_index[0]
  const int* dst = eidx + E;      // edge_index[1]

  // workspace: dinv[N] | bufA[N*16] (h1 then h2) | bufB[N*16] (agg1 -> a1)
  float* dinv = (float*)d_ws;
  float* bufA = dinv + N;
  float* bufB = bufA + (size_t)N * F_H;

  const int TB = 256;
  int nb_n  = (N + TB - 1) / TB;
  int nb_e  = (E + TB - 1) / TB;
  int total_sc = E * F_H;                     // 51.2M, fits int
  int nb_sc = (total_sc + TB - 1) / TB;
  int nb_nf = (N * F_H + TB - 1) / TB;
  int nb_g  = (N + 16 * G_WAVES - 1) / (16 * G_WAVES);

  // normalization
  k_init_deg<<<nb_n, TB, 0, stream>>>(dinv, N);
  k_edge_deg<<<nb_e, TB, 0, stream>>>(dst, dinv, E);
  k_rsqrt<<<nb_n, TB, 0, stream>>>(dinv, N);

  // layer 1
  k_gemm1_wmma<<<nb_g, 32 * G_WAVES, 0, stream>>>(x, W1, dinv, bufA, bufB, N);
  k_scatter<<<nb_sc, TB, 0, stream>>>(bufA, src, dst, dinv, bufB, total_sc);
  k_bias_relu<<<nb_nf, TB, 0, stream>>>(bufB, b1, N * F_H);

  // layer 2 (accumulator lives directly in d_out)
  k_gemm2_wmma<<<nb_g, 32 * G_WAVES, 0, stream>>>(bufB, W2, dinv, bufA, out, N);
  k_scatter<<<nb_sc, TB, 0, stream>>>(bufA, src, dst, dinv, out, total_sc);
  k_bias_logsoftmax<<<nb_n, TB, 0, stream>>>(out, b2, N);
}